// DEPNet_72473278153363
// MI455X (gfx1250) — compile-verified
//
#include <hip/hip_runtime.h>
#include <math.h>

// ---------------- problem constants ----------------
#define Bc   256
#define Dc   1280
#define Nc   49
#define Kc   8
#define KD   (Kc*Dc)        // 10240
#define EPSB 1e-5f
#define KS   8              // split-K factor for enc / fc1 GEMMs

typedef __attribute__((ext_vector_type(2))) float v2f;
typedef __attribute__((ext_vector_type(8))) float v8f;

// ---------------- workspace layout (floats) ----------------
#define OFF_SCALE  ((size_t)0)                       // 1280  bn2 fused scale
#define OFF_SHIFT  (OFF_SCALE + 1280)                // 1280  bn2 fused shift
#define OFF_C2     (OFF_SHIFT + 1280)                // 8 (padded 256) ||C_k||^2
#define OFF_ENORM  (OFF_C2 + 256)                    // 256   ||E_b||^2
#define OFF_P      (OFF_ENORM + 256)                 // B*D   pooled means (post-BN)
#define OFF_A      (OFF_P + (size_t)Bc*Dc)           // B*K*N softmax assignments [b][k][n]
#define OFF_E      (OFF_A + (size_t)Bc*Kc*Nc)        // B*K*D aggregated residuals [b][k*D+d]
#define OFF_X1     (OFF_E + (size_t)Bc*KD)           // B*64  encoding branch
#define OFF_X2     (OFF_X1 + (size_t)Bc*64)          // B*64  pool branch (post BN1d)
#define OFF_OUTER  (OFF_X2 + (size_t)Bc*64)          // B*4096 normalized outer product
#define OFF_H      (OFF_OUTER + (size_t)Bc*4096)     // B*128 fc1 output
#define OFF_PE     (OFF_H + (size_t)Bc*128)          // KS*B*64  enc split-K partials
#define OFF_PH     (OFF_PE + (size_t)KS*Bc*64)       // KS*B*128 fc1 split-K partials

// =====================================================================
// K0: fold BN2d into per-channel scale/shift, precompute ||C_k||^2
// =====================================================================
__global__ void k0_prep(const float* __restrict__ g, const float* __restrict__ be,
                        const float* __restrict__ mu, const float* __restrict__ var,
                        const float* __restrict__ cw, float* __restrict__ ws) {
    const int tid = threadIdx.x;            // 1 block x 256
    for (int d = tid; d < Dc; d += 256) {
        float inv = g[d] / sqrtf(var[d] + EPSB);
        ws[OFF_SCALE + d] = inv;
        ws[OFF_SHIFT + d] = be[d] - mu[d] * inv;
    }
    if (tid < Kc) {
        float s = 0.f;
        for (int d = 0; d < Dc; ++d) { float c = cw[tid*Dc + d]; s += c*c; }
        ws[OFF_C2 + tid] = s;
    }
}

// =====================================================================
// K1: pooled means p[b,d] = scale[d]*mean_n(x) + shift[d]
// =====================================================================
__global__ void k1_pool(const float* __restrict__ x, float* __restrict__ ws) {
    const int idx = blockIdx.x * 256 + threadIdx.x;   // = b*D + d, exact cover
    const int d = idx % Dc;
    const float* xr = x + (size_t)idx * Nc;
    float s = 0.f;
#pragma unroll
    for (int n = 0; n < Nc; ++n) s += xr[n];
    ws[OFF_P + idx] = ws[OFF_SCALE + d] * (s * (1.f / 49.f)) + ws[OFF_SHIFT + d];
}

// =====================================================================
// K2: per (b,n): xc_k = X_bn . C_k , x2 = ||X_bn||^2, softmax over K -> A
// =====================================================================
__global__ void k2_assign(const float* __restrict__ x, const float* __restrict__ cw,
                          const float* __restrict__ sc, float* __restrict__ ws) {
    const int bn = blockIdx.x;              // B*N blocks
    const int b = bn / Nc, n = bn % Nc;
    const int tid = threadIdx.x;            // 256
    float acc[9];
#pragma unroll
    for (int j = 0; j < 9; ++j) acc[j] = 0.f;

    for (int d = tid; d < Dc; d += 256) {
        float xv = x[((size_t)b*Dc + d)*Nc + n] * ws[OFF_SCALE + d] + ws[OFF_SHIFT + d];
        acc[0] += xv * xv;
#pragma unroll
        for (int k = 0; k < Kc; ++k) acc[1+k] += xv * cw[k*Dc + d];
    }
    __shared__ float red[256*9];
#pragma unroll
    for (int j = 0; j < 9; ++j) red[tid*9 + j] = acc[j];
    __syncthreads();
    for (int s = 128; s > 0; s >>= 1) {
        if (tid < s) {
#pragma unroll
            for (int j = 0; j < 9; ++j) red[tid*9 + j] += red[(tid+s)*9 + j];
        }
        __syncthreads();
    }
    if (tid == 0) {
        float x2 = red[0];
        float sl[Kc], m = -3.402823e38f;
#pragma unroll
        for (int k = 0; k < Kc; ++k) {
            sl[k] = sc[k] * (x2 + ws[OFF_C2 + k] - 2.f * red[1+k]);
            m = fmaxf(m, sl[k]);
        }
        float ssum = 0.f;
#pragma unroll
        for (int k = 0; k < Kc; ++k) { float e = __expf(sl[k] - m); sl[k] = e; ssum += e; }
        float inv = 1.f / ssum;
#pragma unroll
        for (int k = 0; k < Kc; ++k)
            ws[OFF_A + ((size_t)b*Kc + k)*Nc + n] = sl[k] * inv;
    }
}

// =====================================================================
// K3: E[b,k,d] = sum_n A[b,k,n]*xb[b,d,n] - asum[b,k]*C[k,d]; also ||E_b||^2
// =====================================================================
__global__ void k3_aggregate(const float* __restrict__ x, const float* __restrict__ cw,
                             float* __restrict__ ws) {
    const int b = blockIdx.x, tid = threadIdx.x;     // 256 threads
    __shared__ float As[Kc*Nc];
    __shared__ float asum[Kc];
    __shared__ float red[256];
    for (int i = tid; i < Kc*Nc; i += 256) As[i] = ws[OFF_A + (size_t)b*Kc*Nc + i];
    __syncthreads();
    if (tid < Kc) {
        float s = 0.f;
        for (int n = 0; n < Nc; ++n) s += As[tid*Nc + n];
        asum[tid] = s;
    }
    __syncthreads();

    float ss = 0.f;
    for (int it = 0; it < 5; ++it) {
        const int d = it*256 + tid;                  // < 1280 always
        const float s = ws[OFF_SCALE + d], t = ws[OFF_SHIFT + d];
        const float* xr = x + ((size_t)b*Dc + d)*Nc;
        float xv[Nc];
#pragma unroll
        for (int n = 0; n < Nc; ++n) xv[n] = xr[n]*s + t;
#pragma unroll
        for (int k = 0; k < Kc; ++k) {
            float e = -asum[k] * cw[k*Dc + d];
#pragma unroll
            for (int n = 0; n < Nc; ++n) e += As[k*Nc + n] * xv[n];
            ws[OFF_E + ((size_t)b*Kc + k)*Dc + d] = e;
            ss += e*e;
        }
    }
    red[tid] = ss; __syncthreads();
    for (int s = 128; s > 0; s >>= 1) {
        if (tid < s) red[tid] += red[tid+s];
        __syncthreads();
    }
    if (tid == 0) ws[OFF_ENORM + b] = red[0];
}

// =====================================================================
// fp32 WMMA 16x16 tile: D += A(16xK) * B(Kx16), exact fp32 FMA semantics.
// A row-major [M,ldk]; W row-major [Nout,ldk] supplies B = W^T tiles.
// A frag: lane l -> M=l&15, VGPR0/1 = K + 2*(l>=16) + {0,1}; B symmetric.
// =====================================================================
__device__ __forceinline__ v8f wmma_f32_tile(const float* __restrict__ a_row,
                                             const float* __restrict__ w_row,
                                             int ldk, int kdim) {
    const int lane = threadIdx.x & 31;
    const int half = lane >> 4;
    const int lm   = lane & 15;
    const float* ap = a_row + (size_t)lm * ldk + 2*half;
    const float* bp = w_row + (size_t)lm * ldk + 2*half;
    v8f acc = {};
#pragma unroll 8
    for (int k = 0; k < kdim; k += 4) {
        v2f av = *(const v2f*)(ap + k);
        v2f bv = *(const v2f*)(bp + k);
        acc = __builtin_amdgcn_wmma_f32_16x16x4_f32(false, av, false, bv,
                                                    (short)0, acc, false, false);
    }
    return acc;
}

// =====================================================================
// K4a: enc GEMM split-K partials  (256 x 64 x 10240, KS=8 chunks of 1280)
//      grid (16, KS), 4 waves per block (one 16-col tile each)
// =====================================================================
__global__ void k4a_enc_partial(const float* __restrict__ w, float* __restrict__ ws) {
    const int m0 = blockIdx.x * 16;
    const int kc = blockIdx.y;                       // K chunk
    const int n0 = (threadIdx.x >> 5) * 16;
    const int kchunk = KD / KS;                      // 1280
    v8f acc = wmma_f32_tile(ws + OFF_E + (size_t)m0*KD + (size_t)kc*kchunk,
                            w + (size_t)n0*KD + (size_t)kc*kchunk, KD, kchunk);
    const int lane = threadIdx.x & 31, half = lane >> 4, lm = lane & 15;
    const int col = n0 + lm;
#pragma unroll
    for (int i = 0; i < 8; ++i) {
        const int row = m0 + i + 8*half;
        ws[OFF_PE + ((size_t)kc*Bc + row)*64 + col] = acc[i];
    }
}

// K4b: sum KS partials in fixed order (deterministic), apply 1/||E|| and bias
__global__ void k4b_enc_reduce(const float* __restrict__ bias, float* __restrict__ ws) {
    const int idx = blockIdx.x * 256 + threadIdx.x;  // 64 blocks -> 256*64
    const int row = idx >> 6, col = idx & 63;
    float s = 0.f;
#pragma unroll
    for (int kc = 0; kc < KS; ++kc) s += ws[OFF_PE + ((size_t)kc*Bc + row)*64 + col];
    const float rn = 1.f / fmaxf(sqrtf(ws[OFF_ENORM + row]), 1e-12f);
    ws[OFF_X1 + (size_t)row*64 + col] = s*rn + bias[col];
}

// =====================================================================
// K5: x2b = BN1d(p @ pool_w^T + pool_b)   (256 x 64 x 1280, fp32 WMMA)
// =====================================================================
__global__ void k5_pool_gemm(const float* __restrict__ w, const float* __restrict__ bias,
                             const float* __restrict__ g1, const float* __restrict__ b1,
                             const float* __restrict__ m1, const float* __restrict__ v1,
                             float* __restrict__ ws) {
    const int m0 = blockIdx.x * 16;
    const int n0 = (threadIdx.x >> 5) * 16;
    v8f acc = wmma_f32_tile(ws + OFF_P + (size_t)m0*Dc, w + (size_t)n0*Dc, Dc, Dc);
    const int lane = threadIdx.x & 31, half = lane >> 4, lm = lane & 15;
    const int col = n0 + lm;
    const float bc = bias[col];
    const float inv = g1[col] / sqrtf(v1[col] + EPSB);
    const float mm = m1[col], bb = b1[col];
#pragma unroll
    for (int i = 0; i < 8; ++i) {
        const int row = m0 + i + 8*half;
        float v = acc[i] + bc;
        ws[OFF_X2 + (size_t)row*64 + col] = (v - mm) * inv + bb;
    }
}

// =====================================================================
// K6: normalized outer product. ||x2 (x) x1||_F = ||x2||*||x1||.
// =====================================================================
__global__ void k6_outer(float* __restrict__ ws) {
    const int b = blockIdx.x, tid = threadIdx.x;     // 256 blocks x 256
    __shared__ float x1s[64], x2s[64], red[256], nn[2];
    if (tid < 64) {
        x1s[tid] = ws[OFF_X1 + (size_t)b*64 + tid];
        x2s[tid] = ws[OFF_X2 + (size_t)b*64 + tid];
    }
    __syncthreads();
    red[tid] = (tid < 64) ? x1s[tid]*x1s[tid] : 0.f;
    __syncthreads();
    for (int s = 128; s > 0; s >>= 1) { if (tid < s) red[tid] += red[tid+s]; __syncthreads(); }
    if (tid == 0) nn[0] = red[0];
    __syncthreads();
    red[tid] = (tid < 64) ? x2s[tid]*x2s[tid] : 0.f;
    __syncthreads();
    for (int s = 128; s > 0; s >>= 1) { if (tid < s) red[tid] += red[tid+s]; __syncthreads(); }
    if (tid == 0) nn[1] = red[0];
    __syncthreads();
    const float rn1 = 1.f / fmaxf(sqrtf(nn[0]), 1e-12f);
    const float rn2 = 1.f / fmaxf(sqrtf(nn[1]), 1e-12f);
    for (int i = tid; i < 4096; i += 256) {
        const int r = i >> 6, c = i & 63;
        ws[OFF_OUTER + (size_t)b*4096 + i] = (x2s[r]*rn2) * (x1s[c]*rn1);
    }
}

// =====================================================================
// K7a: fc1 GEMM split-K partials (256 x 128 x 4096, KS=8 chunks of 512)
//      grid (16, KS), 8 waves per block (one 16-col tile each)
// =====================================================================
__global__ void k7a_fc1_partial(const float* __restrict__ w, float* __restrict__ ws) {
    const int m0 = blockIdx.x * 16;
    const int kc = blockIdx.y;
    const int n0 = (threadIdx.x >> 5) * 16;          // 8 waves -> 128 cols
    const int kchunk = 4096 / KS;                    // 512
    v8f acc = wmma_f32_tile(ws + OFF_OUTER + (size_t)m0*4096 + (size_t)kc*kchunk,
                            w + (size_t)n0*4096 + (size_t)kc*kchunk, 4096, kchunk);
    const int lane = threadIdx.x & 31, half = lane >> 4, lm = lane & 15;
    const int col = n0 + lm;
#pragma unroll
    for (int i = 0; i < 8; ++i) {
        const int row = m0 + i + 8*half;
        ws[OFF_PH + ((size_t)kc*Bc + row)*128 + col] = acc[i];
    }
}

// K7b: deterministic reduction + bias -> h
__global__ void k7b_fc1_reduce(const float* __restrict__ bias, float* __restrict__ ws) {
    const int idx = blockIdx.x * 256 + threadIdx.x;  // 128 blocks -> 256*128
    const int row = idx >> 7, col = idx & 127;
    float s = 0.f;
#pragma unroll
    for (int kc = 0; kc < KS; ++kc) s += ws[OFF_PH + ((size_t)kc*Bc + row)*128 + col];
    ws[OFF_H + (size_t)row*128 + col] = s + bias[col];
}

// =====================================================================
// K8: out = l2norm(h) @ fc2_w^T + fc2_b   (tiny: 23 outputs, K=128)
// =====================================================================
__global__ void k8_fc2(const float* __restrict__ w, const float* __restrict__ bias,
                       const float* __restrict__ ws, float* __restrict__ out) {
    const int b = blockIdx.x, tid = threadIdx.x;     // 256 blocks x 128
    __shared__ float hs[128], red[128], rns;
    hs[tid] = ws[OFF_H + (size_t)b*128 + tid];
    red[tid] = hs[tid]*hs[tid];
    __syncthreads();
    for (int s = 64; s > 0; s >>= 1) { if (tid < s) red[tid] += red[tid+s]; __syncthreads(); }
    if (tid == 0) rns = 1.f / fmaxf(sqrtf(red[0]), 1e-12f);
    __syncthreads();
    if (tid < 23) {
        float s = bias[tid];
        const float r = rns;
        for (int j = 0; j < 128; ++j) s += hs[j] * r * w[tid*128 + j];
        out[b*23 + tid] = s;
    }
}

// =====================================================================
extern "C" void kernel_launch(void* const* d_in, const int* in_sizes, int n_in,
                              void* d_out, int out_size, void* d_ws, size_t ws_size,
                              hipStream_t stream) {
    const float* x        = (const float*)d_in[0];
    const float* bn2_g    = (const float*)d_in[1];
    const float* bn2_b    = (const float*)d_in[2];
    const float* bn2_m    = (const float*)d_in[3];
    const float* bn2_v    = (const float*)d_in[4];
    const float* cw       = (const float*)d_in[5];
    const float* sc       = (const float*)d_in[6];
    const float* enc_w    = (const float*)d_in[7];
    const float* enc_b    = (const float*)d_in[8];
    const float* pool_w   = (const float*)d_in[9];
    const float* pool_b   = (const float*)d_in[10];
    const float* bn1_g    = (const float*)d_in[11];
    const float* bn1_b    = (const float*)d_in[12];
    const float* bn1_m    = (const float*)d_in[13];
    const float* bn1_v    = (const float*)d_in[14];
    const float* fc1_w    = (const float*)d_in[15];
    const float* fc1_b    = (const float*)d_in[16];
    const float* fc2_w    = (const float*)d_in[17];
    const float* fc2_b    = (const float*)d_in[18];
    float* ws  = (float*)d_ws;
    float* out = (float*)d_out;

    k0_prep        <<<1,              256, 0, stream>>>(bn2_g, bn2_b, bn2_m, bn2_v, cw, ws);
    k1_pool        <<<Bc*Dc/256,      256, 0, stream>>>(x, ws);
    k2_assign      <<<Bc*Nc,          256, 0, stream>>>(x, cw, sc, ws);
    k3_aggregate   <<<Bc,             256, 0, stream>>>(x, cw, ws);
    k5_pool_gemm   <<<Bc/16,          128, 0, stream>>>(pool_w, pool_b, bn1_g, bn1_b, bn1_m, bn1_v, ws);
    k4a_enc_partial<<<dim3(Bc/16,KS), 128, 0, stream>>>(enc_w, ws);
    k4b_enc_reduce <<<Bc*64/256,      256, 0, stream>>>(enc_b, ws);
    k6_outer       <<<Bc,             256, 0, stream>>>(ws);
    k7a_fc1_partial<<<dim3(Bc/16,KS), 256, 0, stream>>>(fc1_w, ws);
    k7b_fc1_reduce <<<Bc*128/256,     256, 0, stream>>>(fc1_b, ws);
    k8_fc2         <<<Bc,             128, 0, stream>>>(fc2_w, fc2_b, ws, out);
}